// EMOGI_9294309229064
// MI455X (gfx1250) — compile-verified
//
#include <hip/hip_runtime.h>
#include <hip/hip_bf16.h>

typedef float v2f __attribute__((ext_vector_type(2)));
typedef float v8f __attribute__((ext_vector_type(8)));

constexpr int F_IN = 58;
constexpr int KP1  = 64;    // padded K for layer-1 inputs (stride of xp/tx1p)
constexpr int H1   = 300;
constexpr int HP1  = 304;   // padded layer-1 width (19 tiles * 16)
constexpr int H2   = 100;
constexpr int HP2  = 112;   // padded layer-2 width (7 tiles * 16)
constexpr int NT1  = HP1 / 16;   // 19
constexpr int NT2  = HP2 / 16;   // 7

// ---------------------------------------------------------------------------
// Edge preprocessing
// ---------------------------------------------------------------------------
__global__ void k_prep_edges(const long long* __restrict__ ei, int E,
                             int* __restrict__ src32, int* __restrict__ dst32,
                             float* __restrict__ deg) {
    int i = blockIdx.x * blockDim.x + threadIdx.x;
    if (i >= E) return;
    int s = (int)ei[i];
    int d = (int)ei[(size_t)E + i];
    src32[i] = s;
    dst32[i] = d;
    atomicAdd(&deg[s], 1.0f);
}

__global__ void k_dinv(float* __restrict__ deg, int N) {
    int n = blockIdx.x * blockDim.x + threadIdx.x;
    if (n >= N) return;
    float dg = deg[n];
    deg[n] = dg > 0.0f ? rsqrtf(dg) : 0.0f;   // in place: deg -> d^{-1/2}
}

__global__ void k_edgew(const int* __restrict__ src32, const int* __restrict__ dst32,
                        const float* __restrict__ dinv, float* __restrict__ w, int E) {
    int i = blockIdx.x * blockDim.x + threadIdx.x;
    if (i >= E) return;
    w[i] = -(dinv[src32[i]] * dinv[dst32[i]]);
}

// ---------------------------------------------------------------------------
// Packing kernels (run once per call; trivially cheap)
// ---------------------------------------------------------------------------
// x [N,58] -> xp [N,64] zero-padded
__global__ void k_pad_x(const float* __restrict__ x, float* __restrict__ xp, int N) {
    int idx = blockIdx.x * blockDim.x + threadIdx.x;
    if (idx >= N * KP1) return;
    int n = idx / KP1, k = idx % KP1;
    xp[idx] = (k < F_IN) ? x[(size_t)n * F_IN + k] : 0.0f;
}

// W1 (2,58,300) -> WT1 [2][304][64] transposed+padded: WT1[c][col][k] = W1[c][k][col]
__global__ void k_pack_w1(const float* __restrict__ W1, float* __restrict__ WT1) {
    int idx = blockIdx.x * blockDim.x + threadIdx.x;
    if (idx >= 2 * HP1 * KP1) return;
    int c = idx / (HP1 * KP1);
    int rem = idx % (HP1 * KP1);
    int col = rem / KP1, k = rem % KP1;
    WT1[idx] = (col < H1 && k < F_IN) ? W1[(size_t)c * F_IN * H1 + (size_t)k * H1 + col] : 0.0f;
}

// W2 (2,300,100) -> WT2 [2][112][304]: WT2[c][col][k] = W2[c][k][col]
__global__ void k_pack_w2(const float* __restrict__ W2, float* __restrict__ WT2) {
    int idx = blockIdx.x * blockDim.x + threadIdx.x;
    if (idx >= 2 * HP2 * HP1) return;
    int c = idx / (HP2 * HP1);
    int rem = idx % (HP2 * HP1);
    int col = rem / HP1, k = rem % HP1;
    WT2[idx] = (col < H2 && k < H1) ? W2[(size_t)c * H1 * H2 + (size_t)k * H2 + col] : 0.0f;
}

__global__ void k_pad_bias(const float* __restrict__ b, float* __restrict__ bp,
                           int n, int np) {
    int i = blockIdx.x * blockDim.x + threadIdx.x;
    if (i >= np) return;
    bp[i] = (i < n) ? b[i] : 0.0f;
}

// ---------------------------------------------------------------------------
// Weighted scatter-add: acc[dst,:F] += w[e] * feat[src,:F]  (strided rows)
// ---------------------------------------------------------------------------
__global__ void k_agg(const int* __restrict__ src32, const int* __restrict__ dst32,
                      const float* __restrict__ w, const float* __restrict__ feat,
                      float* __restrict__ acc, int E, int F, int fstride, int astride) {
    int i = blockIdx.x * blockDim.x + threadIdx.x;
    if (i >= E) return;
    int s = src32[i], d = dst32[i];
    float we = w[i];
    const float* fs = feat + (size_t)s * fstride;
    float*       fd = acc  + (size_t)d * astride;
    for (int f = 0; f < F; ++f)
        atomicAdd(&fd[f], we * fs[f]);
}

// ---------------------------------------------------------------------------
// Layer 1 GEMM (dual-WMMA, branch-free inner loop):
//   h[N,304] = relu(xp@W1[0] + tx1p@W1[1] + b1p)   (pad cols come out 0)
// ---------------------------------------------------------------------------
__global__ void k_gemm1(const float* __restrict__ xp, const float* __restrict__ tx1p,
                        const float* __restrict__ WT1, const float* __restrict__ b1p,
                        float* __restrict__ h, int mtiles, int Nnodes) {
    const int lane = threadIdx.x & 31;
    const int wave = threadIdx.x >> 5;
    int t = blockIdx.x * 8 + wave;
    if (t >= mtiles * NT1) return;                 // wave-uniform exit
    int mt = t / NT1;
    int nt = t % NT1;

    int row = mt * 16 + (lane & 15);
    if (row >= Nnodes) row = Nnodes - 1;           // clamp: dup rows feed masked outputs
    const int col   = nt * 16 + (lane & 15);       // always < 304
    const int khalf = (lane >> 4) * 2;             // this lane's K pair base

    const v2f* ax  = (const v2f*)(xp   + (size_t)row * KP1);
    const v2f* at  = (const v2f*)(tx1p + (size_t)row * KP1);
    const v2f* bw0 = (const v2f*)(WT1  + (size_t)col * KP1);
    const v2f* bw1 = (const v2f*)(WT1  + (size_t)(HP1 + col) * KP1);

    v8f c = {};
    #pragma unroll
    for (int k0 = 0; k0 < 60; k0 += 4) {           // covers K=58; pad K is zero
        int kh = (k0 + khalf) >> 1;                // v2f index (aligned 8B)
        v2f a0 = ax[kh];
        v2f a1 = at[kh];
        v2f b0 = bw0[kh];
        v2f b1 = bw1[kh];
        c = __builtin_amdgcn_wmma_f32_16x16x4_f32(false, a0, false, b0, (short)0, c, false, false);
        c = __builtin_amdgcn_wmma_f32_16x16x4_f32(false, a1, false, b1, (short)0, c, false, false);
    }

    float b = b1p[col];
    int rbase = mt * 16 + ((lane >> 4) << 3);      // C/D: M = v + 8*(lane>=16)
    #pragma unroll
    for (int v = 0; v < 8; ++v) {
        int r = rbase + v;
        if (r < Nnodes) {
            float val = c[v] + b;
            h[(size_t)r * HP1 + col] = val > 0.0f ? val : 0.0f;
        }
    }
}

// ---------------------------------------------------------------------------
// Layer 2 GEMM (shared A, two accumulators): y0 = h@W2[0], y1 = h@W2[1]
// K runs over padded 304 (pad contributes 0); out width padded to 112.
// ---------------------------------------------------------------------------
__global__ void k_gemm2(const float* __restrict__ h, const float* __restrict__ WT2,
                        float* __restrict__ y0, float* __restrict__ y1,
                        int mtiles, int Nnodes) {
    const int lane = threadIdx.x & 31;
    const int wave = threadIdx.x >> 5;
    int t = blockIdx.x * 8 + wave;
    if (t >= mtiles * NT2) return;
    int mt = t / NT2;
    int nt = t % NT2;

    int row = mt * 16 + (lane & 15);
    if (row >= Nnodes) row = Nnodes - 1;
    const int col   = nt * 16 + (lane & 15);       // always < 112
    const int khalf = (lane >> 4) * 2;

    const v2f* ah  = (const v2f*)(h   + (size_t)row * HP1);
    const v2f* bw0 = (const v2f*)(WT2 + (size_t)col * HP1);
    const v2f* bw1 = (const v2f*)(WT2 + (size_t)(HP2 + col) * HP1);

    v8f c0 = {};
    v8f c1 = {};
    #pragma unroll 4
    for (int k0 = 0; k0 < HP1; k0 += 4) {          // 76 steps, branch-free
        int kh = (k0 + khalf) >> 1;
        v2f a  = ah[kh];
        v2f b0 = bw0[kh];
        v2f b1 = bw1[kh];
        c0 = __builtin_amdgcn_wmma_f32_16x16x4_f32(false, a, false, b0, (short)0, c0, false, false);
        c1 = __builtin_amdgcn_wmma_f32_16x16x4_f32(false, a, false, b1, (short)0, c1, false, false);
    }

    int rbase = mt * 16 + ((lane >> 4) << 3);
    #pragma unroll
    for (int v = 0; v < 8; ++v) {
        int r = rbase + v;
        if (r < Nnodes) {
            y0[(size_t)r * HP2 + col] = c0[v];
            y1[(size_t)r * HP2 + col] = c1[v];
        }
    }
}

// h2 = relu(y0 + agg2 + b2p), in place into y0 (pad cols stay 0)
__global__ void k_combine2(const float* __restrict__ agg, const float* __restrict__ b2p,
                           float* __restrict__ y0, int total) {
    int idx = blockIdx.x * blockDim.x + threadIdx.x;
    if (idx >= total) return;
    int f = idx % HP2;
    float v = y0[idx] + agg[idx] + b2p[f];
    y0[idx] = v > 0.0f ? v : 0.0f;
}

// Layer 3 projections: z0 = h2@W3[0], z1 = h2@W3[1]  (100 -> 1)
__global__ void k_layer3(const float* __restrict__ h2, const float* __restrict__ W3,
                         float* __restrict__ z0, float* __restrict__ z1, int N) {
    int n = blockIdx.x * blockDim.x + threadIdx.x;
    if (n >= N) return;
    const float* r = h2 + (size_t)n * HP2;
    float s0 = 0.0f, s1 = 0.0f;
    #pragma unroll 4
    for (int k = 0; k < H2; ++k) {
        float hv = r[k];
        s0 += hv * W3[k];
        s1 += hv * W3[H2 + k];
    }
    z0[n] = s0;
    z1[n] = s1;
}

__global__ void k_final(const float* __restrict__ z0, const float* __restrict__ agg3,
                        const float* __restrict__ b3, float* __restrict__ out, int N) {
    int n = blockIdx.x * blockDim.x + threadIdx.x;
    if (n >= N) return;
    out[n] = z0[n] + agg3[n] + b3[0];
}

// ---------------------------------------------------------------------------
extern "C" void kernel_launch(void* const* d_in, const int* in_sizes, int n_in,
                              void* d_out, int out_size, void* d_ws, size_t ws_size,
                              hipStream_t stream) {
    const float*     x   = (const float*)d_in[0];
    const long long* ei  = (const long long*)d_in[1];
    const float*     W1  = (const float*)d_in[2];
    const float*     b1  = (const float*)d_in[3];
    const float*     W2  = (const float*)d_in[4];
    const float*     b2  = (const float*)d_in[5];
    const float*     W3  = (const float*)d_in[6];
    const float*     b3  = (const float*)d_in[7];
    float*           out = (float*)d_out;

    const int N = in_sizes[0] / F_IN;   // 100000
    const int E = in_sizes[1] / 2;      // 800000

    // ---- bump allocator over workspace ----
    char* ws = (char*)d_ws;
    size_t off = 0;
    auto take = [&](size_t bytes) -> void* {
        void* p = ws + off;
        off = (off + bytes + 255) & ~(size_t)255;
        return p;
    };
    int*   src32 = (int*)  take((size_t)E * 4);
    int*   dst32 = (int*)  take((size_t)E * 4);
    float* w     = (float*)take((size_t)E * 4);
    float* dinv  = (float*)take((size_t)N * 4);
    float* xp    = (float*)take((size_t)N * KP1 * 4);
    float* tx1p  = (float*)take((size_t)N * KP1 * 4);
    float* WT1   = (float*)take((size_t)2 * HP1 * KP1 * 4);
    float* b1p   = (float*)take((size_t)HP1 * 4);
    float* h     = (float*)take((size_t)N * HP1 * 4);
    float* WT2   = (float*)take((size_t)2 * HP2 * HP1 * 4);
    float* b2p   = (float*)take((size_t)HP2 * 4);
    float* y0    = (float*)take((size_t)N * HP2 * 4);      // later reused as h2
    float* y1    = (float*)take((size_t)N * HP2 * 4);
    float* agg2  = (float*)take((size_t)N * HP2 * 4);
    float* z0    = (float*)take((size_t)N * 4);
    float* z1    = (float*)take((size_t)N * 4);
    float* agg3  = (float*)take((size_t)N * 4);

    hipMemsetAsync(dinv, 0, (size_t)N * 4, stream);
    hipMemsetAsync(tx1p, 0, (size_t)N * KP1 * 4, stream);
    hipMemsetAsync(agg2, 0, (size_t)N * HP2 * 4, stream);
    hipMemsetAsync(agg3, 0, (size_t)N * 4, stream);

    const int tb = 256;
    const int gE = (E + tb - 1) / tb;
    const int gN = (N + tb - 1) / tb;

    // Preprocess graph + pack operands
    k_prep_edges<<<gE, tb, 0, stream>>>(ei, E, src32, dst32, dinv);
    k_dinv      <<<gN, tb, 0, stream>>>(dinv, N);
    k_edgew     <<<gE, tb, 0, stream>>>(src32, dst32, dinv, w, E);
    k_pad_x     <<<((N * KP1) + tb - 1) / tb, tb, 0, stream>>>(x, xp, N);
    k_pack_w1   <<<((2 * HP1 * KP1) + tb - 1) / tb, tb, 0, stream>>>(W1, WT1);
    k_pack_w2   <<<((2 * HP2 * HP1) + tb - 1) / tb, tb, 0, stream>>>(W2, WT2);
    k_pad_bias  <<<(HP1 + tb - 1) / tb, tb, 0, stream>>>(b1, b1p, H1, HP1);
    k_pad_bias  <<<(HP2 + tb - 1) / tb, tb, 0, stream>>>(b2, b2p, H2, HP2);

    // Layer 1: aggregate in the small dim (58), then fused dual-WMMA GEMM
    k_agg<<<gE, tb, 0, stream>>>(src32, dst32, w, x, tx1p, E, F_IN, F_IN, KP1);
    const int mtiles = (N + 15) / 16;
    k_gemm1<<<(mtiles * NT1 + 7) / 8, 256, 0, stream>>>(xp, tx1p, WT1, b1p, h, mtiles, N);

    // Layer 2: GEMM first (aggregate over 100 instead of 300)
    k_gemm2<<<(mtiles * NT2 + 7) / 8, 256, 0, stream>>>(h, WT2, y0, y1, mtiles, N);
    k_agg<<<gE, tb, 0, stream>>>(src32, dst32, w, y1, agg2, E, H2, HP2, HP2);
    {
        int total = N * HP2;
        k_combine2<<<(total + tb - 1) / tb, tb, 0, stream>>>(agg2, b2p, y0, total);
    }

    // Layer 3: project to scalar first, then aggregate over F=1
    k_layer3<<<gN, tb, 0, stream>>>(y0, W3, z0, z1, N);
    k_agg   <<<gE, tb, 0, stream>>>(src32, dst32, w, z1, agg3, E, 1, 1, 1);
    k_final <<<gN, tb, 0, stream>>>(z0, agg3, b3, out, N);
}